// HyperbolicAttention_89154931130751
// MI455X (gfx1250) — compile-verified
//
#include <hip/hip_runtime.h>
#include <hip/hip_bf16.h>

// ---------------------------------------------------------------------------
// HyperbolicAttention for MI455X (gfx1250, wave32).
// fp32 end-to-end (hyperbolic maps are sensitive near the Poincare ball
// boundary). All GEMMs use native fp32 V_WMMA_F32_16X16X4_F32, one wave per
// 16x64 output strip (4 accumulators sharing each A fragment).
// Mobius scan carries all norms as wave-uniform scalars: exactly ONE shuffle
// reduction (x.y) per serial step.
// ---------------------------------------------------------------------------

typedef float v2f __attribute__((ext_vector_type(2)));
typedef float v8f __attribute__((ext_vector_type(8)));

#define EPS   1e-5f
#define EPS2  1e-10f
#define MAXN  (1.0f - 1e-5f)

static __device__ __forceinline__ v8f wmma4(v2f a, v2f b, v8f c) {
  // (neg_a, A, neg_b, B, c_mod, C, reuse_a, reuse_b)
  return __builtin_amdgcn_wmma_f32_16x16x4_f32(false, a, false, b, (short)0, c,
                                               false, false);
}

// Block-wide sum over 256 threads (8 wave32s): shuffle within wave, LDS across.
static __device__ __forceinline__ float blk_sum(float v, float* sbuf) {
  #pragma unroll
  for (int m = 16; m >= 1; m >>= 1) v += __shfl_xor(v, m, 32);
  const int lane = threadIdx.x & 31, wid = threadIdx.x >> 5;
  __syncthreads();               // protect sbuf from previous call's readers
  if (lane == 0) sbuf[wid] = v;
  __syncthreads();
  float r = sbuf[0];
  #pragma unroll
  for (int w = 1; w < 8; ++w) r += sbuf[w];
  return r;
}

// Problem constants (reference: B=4, S=512, E=512, H=8, D=64)
#define CB 4
#define CS 512
#define CE 512
#define CH 8
#define CD 64

// ---------------------------------------------------------------------------
// Kernel 1: xt = logmap0(project(x)); one 256-thread block per 512-elem row.
// layout 0: x is [B,S,E]. layout 1: x is attended [B,H,S,D] (gather).
// ---------------------------------------------------------------------------
__global__ void k_pre(const float* __restrict__ x, float* __restrict__ xt,
                      int layout) {
  __shared__ float sbuf[8];
  const int row = blockIdx.x, tid = threadIdx.x;
  const int e0 = 2 * tid, e1 = e0 + 1;
  float a, b;
  if (layout == 0) {
    a = x[(size_t)row * CE + e0];
    b = x[(size_t)row * CE + e1];
  } else {
    const int bb = row >> 9, s = row & (CS - 1);
    a = x[((((size_t)bb * CH + (e0 >> 6)) * CS + s) * CD) + (e0 & 63)];
    b = x[((((size_t)bb * CH + (e1 >> 6)) * CS + s) * CD) + (e1 & 63)];
  }
  float s2 = blk_sum(a * a + b * b, sbuf);
  float n = sqrtf(fmaxf(s2, EPS2));
  float sc = (n >= MAXN) ? (MAXN / (n + EPS)) : 1.0f;
  a *= sc; b *= sc; s2 *= sc * sc;
  float n2 = sqrtf(fmaxf(s2, EPS2));
  float sc2 = (n2 >= MAXN) ? (MAXN / (n2 + EPS)) : 1.0f;
  a *= sc2; b *= sc2; s2 *= sc2 * sc2;
  float n3 = sqrtf(fmaxf(s2, EPS2));
  const float lam = 2.0f / (1.0f + EPS);
  float coef = (2.0f / lam) * atanhf(n3) / n3;
  xt[(size_t)row * CE + e0] = coef * a;
  xt[(size_t)row * CE + e1] = coef * b;
}

// ---------------------------------------------------------------------------
// Kernel 2: C[M,N] = A[M,K] @ W[N,K]^T, fp32 WMMA, one wave per 16x64 strip.
// A 16x4 frag: lanes 0-15 M=lane K={0,1}; lanes 16-31 K={2,3}.
// B  4x16 frag: VGPR r = K=r (lanes 0-15) / K=r+2 (lanes 16-31).
// ---------------------------------------------------------------------------
__global__ void k_gemm(const float* __restrict__ A, const float* __restrict__ W,
                       float* __restrict__ C) {
  const int N = CE, K = CE;
  const int wave = threadIdx.x >> 5, lane = threadIdx.x & 31;
  const int wt = blockIdx.x * 8 + wave;           // (M/16)*(N/64)=1024 strips
  const int tilesN4 = N >> 6;                     // 8
  const int tm = wt / tilesN4, tn4 = wt % tilesN4;
  const int half = lane >> 4, r16 = lane & 15;
  const float* aP = A + (size_t)(tm * 16 + r16) * K + half * 2;
  const float* bP = W + (size_t)(tn4 * 64 + r16) * K + half * 2;
  v8f acc0 = {}, acc1 = {}, acc2 = {}, acc3 = {};
  #pragma unroll 4
  for (int k0 = 0; k0 < K; k0 += 4) {
    v2f a  = *(const v2f*)(aP + k0);
    v2f b0 = *(const v2f*)(bP + k0);
    v2f b1 = *(const v2f*)(bP + (size_t)16 * K + k0);
    v2f b2 = *(const v2f*)(bP + (size_t)32 * K + k0);
    v2f b3 = *(const v2f*)(bP + (size_t)48 * K + k0);
    acc0 = wmma4(a, b0, acc0);
    acc1 = wmma4(a, b1, acc1);
    acc2 = wmma4(a, b2, acc2);
    acc3 = wmma4(a, b3, acc3);
  }
  float* cP = C + (size_t)(tm * 16 + half * 8) * N + tn4 * 64 + r16;
  #pragma unroll
  for (int r = 0; r < 8; ++r) {
    cP[(size_t)r * N + 0]  = acc0[r];
    cP[(size_t)r * N + 16] = acc1[r];
    cP[(size_t)r * N + 32] = acc2[r];
    cP[(size_t)r * N + 48] = acc3[r];
  }
}

// ---------------------------------------------------------------------------
// Kernel 3: r = mobius_add(expmap0(t), bias); then
//   mode 0: per-head project + store [B,H,S,D] + projected head norms (Q/K)
//   mode 1: store [B,H,S,D] + raw head norms ||v||^2 (V; feeds scalar yn)
//   mode 2: store rows [B,S,E] (final output)
// Thread tid holds elements 2*tid, 2*tid+1 so wave w == head w.
// ---------------------------------------------------------------------------
__global__ void k_post(const float* __restrict__ t, const float* __restrict__ bias,
                       float* __restrict__ out, float* __restrict__ nrm, int mode) {
  __shared__ float sbuf[8];
  const int row = blockIdx.x, tid = threadIdx.x;
  const int e0 = 2 * tid, e1 = e0 + 1;
  float t0 = t[(size_t)row * CE + e0], t1 = t[(size_t)row * CE + e1];
  float b0 = bias[e0], b1 = bias[e1];

  // expmap0
  float st = blk_sum(t0 * t0 + t1 * t1, sbuf);
  float n = sqrtf(fmaxf(st, EPS2));
  const float lam = 2.0f / (1.0f + EPS);
  float coef = tanhf(lam * n * 0.5f) / n;
  float r0 = coef * t0, r1 = coef * t1;

  float xn = blk_sum(r0 * r0 + r1 * r1, sbuf);
  float xy = blk_sum(r0 * b0 + r1 * b1, sbuf);
  float bn = blk_sum(b0 * b0 + b1 * b1, sbuf);

  #pragma unroll
  for (int p = 0; p < 2; ++p) {   // expmap0 project + mobius re-project of x
    float nr = sqrtf(fmaxf(xn, EPS2));
    float sr = (nr >= MAXN) ? (MAXN / (nr + EPS)) : 1.0f;
    r0 *= sr; r1 *= sr; xn *= sr * sr; xy *= sr;
  }
  float nb = sqrtf(fmaxf(bn, EPS2));
  float sb = (nb >= MAXN) ? (MAXN / (nb + EPS)) : 1.0f;
  b0 *= sb; b1 *= sb; bn *= sb * sb; xy *= sb;

  float Ac = 1.0f + 2.0f * xy + bn;
  float Bc = 1.0f - xn;
  float inv = 1.0f / (1.0f + 2.0f * xy + xn * bn + EPS);
  float o0 = (Ac * r0 + Bc * b0) * inv;
  float o1 = (Ac * r1 + Bc * b1) * inv;

  float so = blk_sum(o0 * o0 + o1 * o1, sbuf);
  float no = sqrtf(fmaxf(so, EPS2));
  float sco = (no >= MAXN) ? (MAXN / (no + EPS)) : 1.0f;
  o0 *= sco; o1 *= sco;

  if (mode == 2) {
    out[(size_t)row * CE + e0] = o0;
    out[(size_t)row * CE + e1] = o1;
    return;
  }
  const int h = tid >> 5, lane = tid & 31;
  const int bb = row >> 9, s = row & (CS - 1);
  float hn = o0 * o0 + o1 * o1;
  #pragma unroll
  for (int m = 16; m >= 1; m >>= 1) hn += __shfl_xor(hn, m, 32);
  float q0 = o0, q1 = o1, hq = hn;
  if (mode == 0) {  // q = project(head); hq = ||q||^2 after projection
    float nh = sqrtf(fmaxf(hn, EPS2));
    float sh = (nh >= MAXN) ? (MAXN / (nh + EPS)) : 1.0f;
    q0 *= sh; q1 *= sh; hq = hn * sh * sh;
  }
  const size_t base = ((((size_t)bb * CH) + h) * CS + s) * CD;
  out[base + 2 * lane]     = q0;
  out[base + 2 * lane + 1] = q1;
  if (nrm != nullptr && lane == 0) nrm[(((size_t)bb * CH) + h) * CS + s] = hq;
}

// ---------------------------------------------------------------------------
// Kernel 4: per (b,h): qk = Q K^T (512x512x64 WMMA) with fused Poincare
// distance epilogue. One wave per 16x64 strip (4 accumulators).
// ---------------------------------------------------------------------------
__global__ void k_scores(const float* __restrict__ Qh, const float* __restrict__ Kh,
                         const float* __restrict__ qn, const float* __restrict__ kn,
                         float* __restrict__ attn) {
  const int wave = threadIdx.x >> 5, lane = threadIdx.x & 31;
  const int wt = blockIdx.x * 8 + wave;   // 32 mats * (32 tm * 8 tn4) = 8192
  const int bh = wt >> 8, rem = wt & 255;
  const int tm = rem >> 3, tn4 = rem & 7;
  const int half = lane >> 4, r16 = lane & 15;
  const float* aP = Qh + ((size_t)bh * CS + tm * 16 + r16) * CD + half * 2;
  const float* bP = Kh + ((size_t)bh * CS + tn4 * 64 + r16) * CD + half * 2;
  v8f acc0 = {}, acc1 = {}, acc2 = {}, acc3 = {};
  #pragma unroll
  for (int k0 = 0; k0 < CD; k0 += 4) {
    v2f a  = *(const v2f*)(aP + k0);
    v2f b0 = *(const v2f*)(bP + k0);
    v2f b1 = *(const v2f*)(bP + 16 * CD + k0);
    v2f b2 = *(const v2f*)(bP + 32 * CD + k0);
    v2f b3 = *(const v2f*)(bP + 48 * CD + k0);
    acc0 = wmma4(a, b0, acc0);
    acc1 = wmma4(a, b1, acc1);
    acc2 = wmma4(a, b2, acc2);
    acc3 = wmma4(a, b3, acc3);
  }
  const int nBase = tn4 * 64 + r16;
  float knn[4];
  #pragma unroll
  for (int q = 0; q < 4; ++q) knn[q] = kn[(size_t)bh * CS + nBase + q * 16];
  float* aout = attn + (((size_t)bh * CS + tm * 16 + half * 8) * CS) + nBase;
  #pragma unroll
  for (int r = 0; r < 8; ++r) {
    const float qnn = qn[(size_t)bh * CS + tm * 16 + r + half * 8];
    const v8f* accs[4] = {&acc0, &acc1, &acc2, &acc3};
    #pragma unroll
    for (int q = 0; q < 4; ++q) {
      float qk = (*accs[q])[r];
      float num = fmaxf(qnn + knn[q] - 2.0f * qk, 0.0f);
      float den = fmaxf((1.0f - qnn) * (1.0f - knn[q]), EPS);
      float dist = acoshf(1.0f + 2.0f * num / den + EPS);
      aout[(size_t)r * CS + q * 16] = -dist;
    }
  }
}

// ---------------------------------------------------------------------------
// Kernel 5: row softmax over Sk=512; one wave per row.
// ---------------------------------------------------------------------------
__global__ void k_softmax(float* __restrict__ attn) {
  const int wave = threadIdx.x >> 5, lane = threadIdx.x & 31;
  const size_t rowId = (size_t)blockIdx.x * 8 + wave;
  float* p = attn + rowId * CS;
  float v[16];
  float mx = -INFINITY;
  #pragma unroll
  for (int i = 0; i < 16; ++i) { v[i] = p[lane + i * 32]; mx = fmaxf(mx, v[i]); }
  #pragma unroll
  for (int m = 16; m >= 1; m >>= 1) mx = fmaxf(mx, __shfl_xor(mx, m, 32));
  float s = 0.0f;
  #pragma unroll
  for (int i = 0; i < 16; ++i) { v[i] = expf(v[i] - mx); s += v[i]; }
  #pragma unroll
  for (int m = 16; m >= 1; m >>= 1) s += __shfl_xor(s, m, 32);
  const float inv = 1.0f / s;
  #pragma unroll
  for (int i = 0; i < 16; ++i) p[lane + i * 32] = v[i] * inv;
}

// ---------------------------------------------------------------------------
// Kernel 6: sequential Mobius-add scan over 512 key positions.
// One wave per chain; lane holds d=lane, d=lane+32. All norms are carried /
// derived as wave-uniform scalars; only x.y needs a shuffle reduction:
//   yn = wj^2 * vn[j]                       (vn precomputed in k_post)
//   rn = (A^2*xn + 2AB*xy + B^2*yn)*inv^2   (scalar identity)
//   xn carried across iterations.
// ---------------------------------------------------------------------------
__global__ void k_scan(const float* __restrict__ attn, const float* __restrict__ Vh,
                       const float* __restrict__ vn, float* __restrict__ out) {
  const int wave = threadIdx.x >> 5, lane = threadIdx.x & 31;
  const size_t chain = (size_t)blockIdx.x * 8 + wave;   // bh*S + q
  const size_t bh = chain >> 9;
  const float* arow = attn + chain * CS;
  const float* vbase = Vh + bh * CS * CD;
  const float* vnrow = vn + bh * CS;
  float x0 = 0.0f, x1 = 0.0f, xn = 0.0f;
  for (int j = 0; j < CS; ++j) {
    const float wj = arow[j];                       // wave-uniform
    float y0 = wj * vbase[j * CD + lane];
    float y1 = wj * vbase[j * CD + lane + 32];
    float yn = wj * wj * vnrow[j];                  // scalar ||y||^2
    if (j + 8 < CS)
      __builtin_prefetch(vbase + (j + 8) * CD + lane, 0, 1);
    // project(x) (scalar-tracked), project(y)
    float nx = sqrtf(fmaxf(xn, EPS2));
    float sx = (nx >= MAXN) ? (MAXN / (nx + EPS)) : 1.0f;
    x0 *= sx; x1 *= sx; xn *= sx * sx;
    float ny = sqrtf(fmaxf(yn, EPS2));
    float sy = (ny >= MAXN) ? (MAXN / (ny + EPS)) : 1.0f;
    y0 *= sy; y1 *= sy; yn *= sy * sy;
    // the one reduction: xy
    float xy = x0 * y0 + x1 * y1;
    #pragma unroll
    for (int m = 16; m >= 1; m >>= 1) xy += __shfl_xor(xy, m, 32);
    // mobius add
    float Ac = 1.0f + 2.0f * xy + yn;
    float Bc = 1.0f - xn;
    float inv = 1.0f / (1.0f + 2.0f * xy + xn * yn + EPS);
    float r0 = (Ac * x0 + Bc * y0) * inv;
    float r1 = (Ac * x1 + Bc * y1) * inv;
    float rn = (Ac * Ac * xn + 2.0f * Ac * Bc * xy + Bc * Bc * yn) * inv * inv;
    // project(r) (scalar)
    float nr = sqrtf(fmaxf(rn, EPS2));
    float sr = (nr >= MAXN) ? (MAXN / (nr + EPS)) : 1.0f;
    x0 = r0 * sr; x1 = r1 * sr; xn = rn * sr * sr;
  }
  float* o = out + chain * CD;
  o[lane] = x0;
  o[lane + 32] = x1;
}

// ---------------------------------------------------------------------------
// Launch. Workspace layout (floats), requires ws_size >= ~60 MB:
//   xt(1M) t(1M) Qh(1M) Kh(1M) Vh(1M) attended(1M) qn/kn/vn(16K each) attn(8M)
// ---------------------------------------------------------------------------
extern "C" void kernel_launch(void* const* d_in, const int* in_sizes, int n_in,
                              void* d_out, int out_size, void* d_ws, size_t ws_size,
                              hipStream_t stream) {
  const float* q  = (const float*)d_in[0];
  const float* k  = (const float*)d_in[1];
  const float* v  = (const float*)d_in[2];
  const float* Wq = (const float*)d_in[3];
  const float* bq = (const float*)d_in[4];
  const float* Wk = (const float*)d_in[5];
  const float* bk = (const float*)d_in[6];
  const float* Wv = (const float*)d_in[7];
  const float* bv = (const float*)d_in[8];
  const float* Wo = (const float*)d_in[9];
  const float* bo = (const float*)d_in[10];
  float* out = (float*)d_out;

  float* ws = (float*)d_ws;
  const size_t MEG = (size_t)1 << 20;
  float* xt  = ws;
  float* t   = ws + 1 * MEG;
  float* Qh  = ws + 2 * MEG;
  float* Kh  = ws + 3 * MEG;
  float* Vh  = ws + 4 * MEG;
  float* att = ws + 5 * MEG;
  float* qn  = ws + 6 * MEG;
  float* kn  = qn + (1 << 14);
  float* vn  = kn + (1 << 14);
  float* attnM = ws + 7 * MEG;                    // [B,H,S,S]

  const int ROWS = CB * CS;                                   // 2048
  const int GEMM_BLOCKS = (ROWS / 16) * (CE / 64) / 8;        // 128
  const int SCORE_BLOCKS = (CB * CH) * (CS / 16) * (CS / 64) / 8; // 1024
  const int ROWWAVE_BLOCKS = (CB * CH * CS) / 8;              // 2048

  k_pre<<<ROWS, 256, 0, stream>>>(q, xt, 0);
  k_gemm<<<GEMM_BLOCKS, 256, 0, stream>>>(xt, Wq, t);
  k_post<<<ROWS, 256, 0, stream>>>(t, bq, Qh, qn, 0);

  k_pre<<<ROWS, 256, 0, stream>>>(k, xt, 0);
  k_gemm<<<GEMM_BLOCKS, 256, 0, stream>>>(xt, Wk, t);
  k_post<<<ROWS, 256, 0, stream>>>(t, bk, Kh, kn, 0);

  k_pre<<<ROWS, 256, 0, stream>>>(v, xt, 0);
  k_gemm<<<GEMM_BLOCKS, 256, 0, stream>>>(xt, Wv, t);
  k_post<<<ROWS, 256, 0, stream>>>(t, bv, Vh, vn, 1);

  k_scores<<<SCORE_BLOCKS, 256, 0, stream>>>(Qh, Kh, qn, kn, attnM);
  k_softmax<<<ROWWAVE_BLOCKS, 256, 0, stream>>>(attnM);
  k_scan<<<ROWWAVE_BLOCKS, 256, 0, stream>>>(attnM, Vh, vn, att);

  k_pre<<<ROWS, 256, 0, stream>>>(att, xt, 1);
  k_gemm<<<GEMM_BLOCKS, 256, 0, stream>>>(xt, Wo, t);
  k_post<<<ROWS, 256, 0, stream>>>(t, bo, out, nullptr, 2);

  (void)in_sizes; (void)n_in; (void)out_size; (void)ws_size;
}